// VanillaRateRNN_84868553769010
// MI455X (gfx1250) — compile-verified
//
#include <hip/hip_runtime.h>
#include <hip/hip_bf16.h>

typedef __attribute__((ext_vector_type(2))) float v2f;
typedef __attribute__((ext_vector_type(8))) float v8f;

// ---------------------------------------------------------------------------
// Generic   Y[m,n] = sum_k X[m,k] * W[n,k] + bias[n]
// One wave computes one 16x16 tile with V_WMMA_F32_16X16X4_F32 over K.
// A-frag layout (16x4 f32): lanes 0-15 -> M=0..15, K = k0 + {0,1}
//                           lanes 16-31 -> M=0..15, K = k0 + {2,3}
// C/D layout: lane&15 -> N, VGPR r -> M = r + 8*(lane>>4)
// ---------------------------------------------------------------------------
template <int K, int NTILES_N>
__global__ __launch_bounds__(256) void gemm_xWT_bias(
    const float* __restrict__ X, const float* __restrict__ W,
    const float* __restrict__ bias, float* __restrict__ Y, int N)
{
    const int lane  = threadIdx.x & 31;
    const int w     = blockIdx.x * 8 + (threadIdx.x >> 5);
    const int mtile = w / NTILES_N;
    const int ntile = w % NTILES_N;
    const int hi    = lane >> 4;
    const int lm    = lane & 15;

    const float* ap = X + (size_t)(mtile * 16 + lm) * K + (hi << 1);
    const float* bp = W + (size_t)(ntile * 16 + lm) * K + (hi << 1);

    v8f c = {};
#pragma unroll 8
    for (int kb = 0; kb < K / 4; ++kb) {
        v2f a = {ap[0], ap[1]};
        v2f b = {bp[0], bp[1]};
        ap += 4; bp += 4;
        c = __builtin_amdgcn_wmma_f32_16x16x4_f32(
                /*neg_a=*/false, a, /*neg_b=*/false, b,
                /*c_mod=*/(short)0, c, /*reuse_a=*/false, /*reuse_b=*/false);
    }

    const int   n  = ntile * 16 + lm;
    const float bv = bias[n];
#pragma unroll
    for (int r = 0; r < 8; ++r) {
        const int m = mtile * 16 + r + 8 * hi;
        Y[(size_t)m * N + n] = c[r] + bv;
    }
}

// ---------------------------------------------------------------------------
// Recurrent scan.  One workgroup (16 waves = 512 threads) owns 16 batch rows
// for the whole sequence; wave w owns hidden columns [16w, 16w+16).
//   pre  = xp[t] + h_{t-1} * W_hh^T        (xp lives in the h_all slot)
//   h_t  = 0.9*h_{t-1} + 0.1*tanh(pre)     (overwrites xp[t] in place)
// Critical-path optimizations:
//   * W_hh fragments resident in registers (64 x v2f per wave)
//   * h shared through a DOUBLE-BUFFERED padded LDS tile -> 1 barrier/step
//   * xp[t+1] prefetched into registers while step t's WMMA chain runs
// ---------------------------------------------------------------------------
#define LDH 258

__global__ __launch_bounds__(512) void rnn_scan(
    const float* __restrict__ Whh, const float* __restrict__ h0,
    float* __restrict__ hall, int T)
{
    __shared__ float hlds[2][16 * LDH];

    const int lane  = threadIdx.x & 31;
    const int wave  = threadIdx.x >> 5;     // 0..15 : N-tile of hidden dim
    const int btile = blockIdx.x;           // batch rows [16*btile, +16)
    const int hi    = lane >> 4;
    const int lm    = lane & 15;
    const int nc    = wave * 16 + lm;       // this lane's output column

    // Preload W_hh B-fragments for this wave's 16 columns (K = 256).
    v2f bfrag[64];
    {
        const float* wp = Whh + (size_t)nc * 256 + (hi << 1);
#pragma unroll
        for (int kb = 0; kb < 64; ++kb) {
            bfrag[kb] = v2f{wp[0], wp[1]};
            wp += 4;
        }
    }

    // Per-lane base pointers into the xp / h_all stream (row stride = T*256).
    // Element (m, nc) of timestep t lives at ((btile*16+m)*T + t)*256 + nc.
    const size_t rowStride = (size_t)T * 256;
    float*       hp[8];
#pragma unroll
    for (int r = 0; r < 8; ++r) {
        const int m = r + 8 * hi;
        hp[r] = hall + (size_t)(btile * 16 + m) * rowStride + nc;
    }

    // Initialize h (registers mirror the C/D tile layout) and LDS buffer 0.
    float hreg[8];
#pragma unroll
    for (int r = 0; r < 8; ++r) {
        const int m = r + 8 * hi;
        hreg[r] = h0[(size_t)(btile * 16 + m) * 256 + nc];
        hlds[0][m * LDH + nc] = hreg[r];
    }

    // Prefetch xp tile for t = 0.
    float cin[8];
#pragma unroll
    for (int r = 0; r < 8; ++r) cin[r] = hp[r][0];

    __syncthreads();

    for (int t = 0; t < T; ++t) {
        const int buf  = t & 1;
        const int nbuf = buf ^ 1;

        // Consume the prefetched xp tile.
        v8f c;
#pragma unroll
        for (int r = 0; r < 8; ++r) c[r] = cin[r];

        // Immediately issue next step's xp loads (clamped; wave-uniform, so
        // EXEC stays all-ones for the WMMAs below).  Latency overlaps the
        // 64-WMMA chain + tanh of this step.
        const size_t tn = (size_t)((t + 1 < T) ? (t + 1) : (T - 1)) * 256;
#pragma unroll
        for (int r = 0; r < 8; ++r) cin[r] = hp[r][tn];

        // pre += h_{t-1} * W_hh^T   (K = 256 -> 64 f32 WMMAs)
#pragma unroll
        for (int kb = 0; kb < 64; ++kb) {
            const float* p = &hlds[buf][lm * LDH + kb * 4 + (hi << 1)];
            v2f a = {p[0], p[1]};
            c = __builtin_amdgcn_wmma_f32_16x16x4_f32(
                    false, a, false, bfrag[kb], (short)0, c, false, false);
        }

        // Leaky update, then publish h_t to the *other* LDS buffer (no race
        // with this step's readers of `buf`) and to global h_all.
#pragma unroll
        for (int r = 0; r < 8; ++r) {
            const float hn = 0.9f * hreg[r] + 0.1f * tanhf(c[r]);
            const int   m  = r + 8 * hi;
            hlds[nbuf][m * LDH + nc] = hn;
            hp[r][(size_t)t * 256]   = hn;
            hreg[r] = hn;
        }

        __syncthreads();   // h_t visible in hlds[nbuf] before next step reads
    }
}

// ---------------------------------------------------------------------------
// Launch:  inputs  {x, h0, W_xh_w, W_xh_b, W_hh_w, W_hy_w, W_hy_b}
//          outputs { y (B*T*512) , h_all (B*T*256) } concatenated in d_out
// ---------------------------------------------------------------------------
extern "C" void kernel_launch(void* const* d_in, const int* in_sizes, int n_in,
                              void* d_out, int out_size, void* d_ws, size_t ws_size,
                              hipStream_t stream)
{
    const float* x    = (const float*)d_in[0];
    const float* h0   = (const float*)d_in[1];
    const float* Wxh  = (const float*)d_in[2];
    const float* bxh  = (const float*)d_in[3];
    const float* Whh  = (const float*)d_in[4];
    const float* Why  = (const float*)d_in[5];
    const float* bhy  = (const float*)d_in[6];

    const int B  = in_sizes[1] / 256;            // h0 is (B,256)
    const int T  = in_sizes[0] / (B * 128);      // x  is (B,T,128)
    const long long BT = (long long)B * T;

    float* y    = (float*)d_out;                 // (B,T,512)
    float* hall = (float*)d_out + BT * 512;      // (B,T,256): xp, then h_all

    // 1) xp = x * W_xh^T + b_xh   -> staged into the h_all region
    {
        const long long waves  = (BT / 16) * 16; // 16 N-tiles of 256
        const int       blocks = (int)(waves / 8);
        gemm_xWT_bias<128, 16><<<blocks, 256, 0, stream>>>(x, Wxh, bxh, hall, 256);
    }

    // 2) sequential scan, batch-partitioned (in-place xp -> h_all)
    rnn_scan<<<B / 16, 512, 0, stream>>>(Whh, h0, hall, T);

    // 3) y = h_all * W_hy^T + b_hy
    {
        const long long waves  = (BT / 16) * 32; // 32 N-tiles of 512
        const int       blocks = (int)(waves / 8);
        gemm_xWT_bias<256, 32><<<blocks, 256, 0, stream>>>(hall, Why, bhy, y, 512);
    }
}